// GlobalGNN_55765855371427
// MI455X (gfx1250) — compile-verified
//
#include <hip/hip_runtime.h>
#include <math.h>

#define H 128

typedef float v2f __attribute__((ext_vector_type(2)));
typedef float v8f __attribute__((ext_vector_type(8)));
typedef unsigned int u32x4 __attribute__((ext_vector_type(4)));
typedef int i32x4 __attribute__((ext_vector_type(4)));
typedef int i32x8 __attribute__((ext_vector_type(8)));

// ---------------------------------------------------------------------------
// Tensor Data Mover: DMA a rows x cols f32 tile (tile == tensor, row-major,
// stride == cols) from global memory into LDS. Issue from ONE wave, then
// s_wait_tensorcnt 0 + workgroup barrier before consumers touch LDS.
// D# layout per CDNA5 ISA (2-D tensor, higher groups zero).
// This toolchain's builtin is the 6-arg form:
//   (uint32x4 g0, int32x8 g1, int32x4, int32x4, int32x8, i32 cpol)
__device__ __forceinline__ void tdm_load_2d_to_lds(const float* gptr, void* ldsPtr,
                                                   int rows, int cols) {
  const unsigned long long ga = (unsigned long long)(uintptr_t)gptr;
  const unsigned int la = (unsigned int)(uintptr_t)ldsPtr;  // flat low 32 = LDS addr

  u32x4 g0;
  g0.x = 1u;                                   // count=1 (valid user descriptor)
  g0.y = la;                                   // lds_addr (bytes)
  g0.z = (unsigned int)ga;                     // global_addr[31:0]
  g0.w = (unsigned int)((ga >> 32) & 0x01FFFFFFu) | (2u << 30);  // addr[56:32]|type=2

  i32x8 g1;
  g1[0] = 2 << 16;                             // data_size = 2 (4 bytes); mask/flags 0
  g1[1] = (cols & 0xFFFF) << 16;               // tensor_dim0[15:0]
  g1[2] = (rows & 0xFFFF) << 16;               // tensor_dim0[31:16]=0 | tensor_dim1[15:0]
  g1[3] = (cols & 0xFFFF) << 16;               // tensor_dim1[31:16]=0 | tile_dim0
  g1[4] = rows & 0xFFFF;                       // tile_dim1 | tile_dim2=0 (2-D)
  g1[5] = cols;                                // tensor_dim0_stride[31:0]
  g1[6] = 0;                                   // stride0[47:32] | stride1[15:0] (unused)
  g1[7] = 0;                                   // stride1[47:16] (unused)

  i32x4 gz4 = {0, 0, 0, 0};
  i32x8 gz8 = {0, 0, 0, 0, 0, 0, 0, 0};
  __builtin_amdgcn_tensor_load_to_lds(g0, g1, gz4, gz4, gz8, 0);
}

// ---------------------------------------------------------------------------
// zero-fill scratch
__global__ void zero_f32(float* __restrict__ p, size_t n) {
  size_t i = (size_t)blockIdx.x * blockDim.x + threadIdx.x;
  size_t stride = (size_t)gridDim.x * blockDim.x;
  for (; i < n; i += stride) p[i] = 0.0f;
}

// ---------------------------------------------------------------------------
// deg[t] += attr[e_id[e]]  (only targets < n2 can appear)
__global__ void deg_accum(const int* __restrict__ tgt, const int* __restrict__ eid,
                          const float* __restrict__ attr, float* __restrict__ deg,
                          int E, int n2) {
  int i = blockIdx.x * blockDim.x + threadIdx.x;
  if (i >= E) return;
  int t = tgt[i];
  if (t < n2) atomicAdd(deg + t, attr[eid[i]]);
}

// dinv[i] = rsqrt(deg[i] + 1)
__global__ void dinv_k(const float* __restrict__ deg, float* __restrict__ dinv, int n2) {
  int i = blockIdx.x * blockDim.x + threadIdx.x;
  if (i >= n2) return;
  dinv[i] = rsqrtf(deg[i] + 1.0f);
}

// ---------------------------------------------------------------------------
// xW[0:M] = x[0:M] @ W^T  via V_WMMA_F32_16X16X4_F32
// block = 256 threads = 8 waves; block handles 16 rows x 128 cols (wave -> 16-col tile)
// W (64 KB) staged into LDS by the Tensor Data Mover.
__global__ void xw_gemm(const float* __restrict__ x, const float* __restrict__ W,
                        float* __restrict__ xw) {
  __shared__ float Wlds[H * H];  // 64 KB: full weight matrix, row-major W[n][k]

  if ((threadIdx.x >> 5) == 0) {           // wave 0 issues the DMA descriptor
    tdm_load_2d_to_lds(W, Wlds, H, H);
    __builtin_amdgcn_s_wait_tensorcnt(0);
  }
  __syncthreads();

  const int wave = threadIdx.x >> 5;    // 0..7 -> column tile
  const int lane = threadIdx.x & 31;
  const int lm = lane & 15;
  const int hi = lane >> 4;
  const int row0 = blockIdx.x * 16;
  const int col0 = wave * 16;

  const float* __restrict__ arow = x + (size_t)(row0 + lm) * H;  // A row (M = lm)
  const float* __restrict__ brow = Wlds + (col0 + lm) * H;       // B col: B[k][n]=W[n][k]

  v8f acc = {};
#pragma unroll
  for (int k0 = 0; k0 < H; k0 += 4) {
    const int kk = k0 + 2 * hi;
    v2f a = {arow[kk], arow[kk + 1]};
    v2f b = {brow[kk], brow[kk + 1]};
    acc = __builtin_amdgcn_wmma_f32_16x16x4_f32(false, a, false, b, (short)0, acc,
                                                false, false);
  }
  float* __restrict__ crow = xw + (size_t)row0 * H + col0;
#pragma unroll
  for (int r = 0; r < 8; ++r)
    crow[(size_t)(r + 8 * hi) * H + lm] = acc[r];
}

// ---------------------------------------------------------------------------
// GCN edge scatter: agg[tgt] += xW[src] * (dinv[src]*w*dinv[tgt]) ; one wave per edge
__global__ void gcn_scatter(const float* __restrict__ xw, const int* __restrict__ src,
                            const int* __restrict__ tgt, const int* __restrict__ eid,
                            const float* __restrict__ attr, const float* __restrict__ dinv,
                            float* __restrict__ agg, int E, int n2) {
  int wid = (blockIdx.x * blockDim.x + threadIdx.x) >> 5;
  int lane = threadIdx.x & 31;
  if (wid >= E) return;
  int s = src[wid], t = tgt[wid];
  if (t >= n2) return;
  float w = attr[eid[wid]];
  float ds = (s < n2) ? dinv[s] : 1.0f;
  float nrm = ds * w * dinv[t];
  const float4 v = ((const float4*)(xw + (size_t)s * H))[lane];
  float* trow = agg + (size_t)t * H + lane * 4;
  atomicAdd(trow + 0, v.x * nrm);
  atomicAdd(trow + 1, v.y * nrm);
  atomicAdd(trow + 2, v.z * nrm);
  atomicAdd(trow + 3, v.w * nrm);
}

// g1 = agg + xW * dinv^2 + b   (in place over the xW buffer, rows [0, M))
__global__ void finalize_g1(float* __restrict__ xw_g1, const float* __restrict__ agg,
                            const float* __restrict__ dinv, const float* __restrict__ b,
                            int M, int n2) {
  size_t i = (size_t)blockIdx.x * blockDim.x + threadIdx.x;
  if (i >= (size_t)M * H) return;
  int row = (int)(i >> 7);
  int col = (int)(i & 127);
  float di = (row < n2) ? dinv[row] : 1.0f;
  float v = xw_g1[i] * (di * di) + b[col];
  if (row < n2) v += agg[i];
  xw_g1[i] = v;
}

// SAGE edge scatter: ssum[tgt] += g1[src]; cnt[tgt] += 1 ; one wave per edge
__global__ void sage_scatter(const float* __restrict__ g1, const int* __restrict__ src,
                             const int* __restrict__ tgt, float* __restrict__ ssum,
                             float* __restrict__ cnt, int E, int n2) {
  int wid = (blockIdx.x * blockDim.x + threadIdx.x) >> 5;
  int lane = threadIdx.x & 31;
  if (wid >= E) return;
  int s = src[wid], t = tgt[wid];
  if (t >= n2) return;
  const float4 v = ((const float4*)(g1 + (size_t)s * H))[lane];
  float* trow = ssum + (size_t)t * H + lane * 4;
  atomicAdd(trow + 0, v.x);
  atomicAdd(trow + 1, v.y);
  atomicAdd(trow + 2, v.z);
  atomicAdd(trow + 3, v.w);
  if (lane == 0) atomicAdd(cnt + t, 1.0f);
}

// ---------------------------------------------------------------------------
// out = (ssum/max(cnt,1)) @ Wl^T + bl + g1[:n2] @ Wr^T    (pre-normalization)
// block = 16 rows x 128 cols, 8 waves; Wl then Wr staged via TDM through 64KB LDS.
__global__ void out_gemm(const float* __restrict__ ssum, const float* __restrict__ cnt,
                         const float* __restrict__ g1, const float* __restrict__ Wl,
                         const float* __restrict__ bl, const float* __restrict__ Wr,
                         float* __restrict__ outp) {
  __shared__ float Wlds[H * H];  // 64 KB, reused for Wl then Wr

  const int wave = threadIdx.x >> 5;
  const int lane = threadIdx.x & 31;
  const int lm = lane & 15;
  const int hi = lane >> 4;
  const int row0 = blockIdx.x * 16;
  const int col0 = wave * 16;
  const int arow = row0 + lm;

  v8f acc = {};

  // pass 1: A = mean = ssum * rc, B = Wl^T
  if (wave == 0) {
    tdm_load_2d_to_lds(Wl, Wlds, H, H);
    __builtin_amdgcn_s_wait_tensorcnt(0);
  }
  __syncthreads();
  {
    const float rc = 1.0f / fmaxf(cnt[arow], 1.0f);
    const float* __restrict__ ap = ssum + (size_t)arow * H;
    const float* __restrict__ bp = Wlds + (col0 + lm) * H;
#pragma unroll
    for (int k0 = 0; k0 < H; k0 += 4) {
      const int kk = k0 + 2 * hi;
      v2f a = {ap[kk] * rc, ap[kk + 1] * rc};
      v2f b = {bp[kk], bp[kk + 1]};
      acc = __builtin_amdgcn_wmma_f32_16x16x4_f32(false, a, false, b, (short)0, acc,
                                                  false, false);
    }
  }
  __syncthreads();  // all waves done reading Wl before the next DMA overwrites LDS

  // pass 2: A = g1 row, B = Wr^T
  if (wave == 0) {
    tdm_load_2d_to_lds(Wr, Wlds, H, H);
    __builtin_amdgcn_s_wait_tensorcnt(0);
  }
  __syncthreads();
  {
    const float* __restrict__ ap = g1 + (size_t)arow * H;
    const float* __restrict__ bp = Wlds + (col0 + lm) * H;
#pragma unroll
    for (int k0 = 0; k0 < H; k0 += 4) {
      const int kk = k0 + 2 * hi;
      v2f a = {ap[kk], ap[kk + 1]};
      v2f b = {bp[kk], bp[kk + 1]};
      acc = __builtin_amdgcn_wmma_f32_16x16x4_f32(false, a, false, b, (short)0, acc,
                                                  false, false);
    }
  }

  const float bias = bl[col0 + lm];
  float* __restrict__ crow = outp + (size_t)row0 * H + col0;
#pragma unroll
  for (int r = 0; r < 8; ++r)
    crow[(size_t)(r + 8 * hi) * H + lm] = acc[r] + bias;
}

// row L2-normalize in place: one wave per 128-float row
__global__ void l2norm(float* __restrict__ outp, int n2) {
  int wid = (blockIdx.x * blockDim.x + threadIdx.x) >> 5;
  int lane = threadIdx.x & 31;
  if (wid >= n2) return;
  float4* row = (float4*)(outp + (size_t)wid * H);
  float4 v = row[lane];
  float ss = v.x * v.x + v.y * v.y + v.z * v.z + v.w * v.w;
#pragma unroll
  for (int off = 16; off > 0; off >>= 1) ss += __shfl_xor(ss, off, 32);
  float inv = 1.0f / fmaxf(sqrtf(ss), 1e-12f);
  v.x *= inv; v.y *= inv; v.z *= inv; v.w *= inv;
  row[lane] = v;
}

// ---------------------------------------------------------------------------
extern "C" void kernel_launch(void* const* d_in, const int* in_sizes, int n_in,
                              void* d_out, int out_size, void* d_ws, size_t ws_size,
                              hipStream_t stream) {
  const float* x     = (const float*)d_in[0];
  const float* attr  = (const float*)d_in[1];
  const int*   e_id1 = (const int*)d_in[3];
  const int*   src1  = (const int*)d_in[6];
  const int*   tgt1  = (const int*)d_in[7];
  const float* gcn_W = (const float*)d_in[10];
  const float* gcn_b = (const float*)d_in[11];
  const float* Wl1   = (const float*)d_in[15];
  const float* bl1   = (const float*)d_in[16];
  const float* Wr1   = (const float*)d_in[17];

  const int E1 = in_sizes[6];
  const int n1 = 50000;  // N1 (host scalar lives on device; fixed by the reference)
  const int n2 = 10000;  // N2

  float* ws   = (float*)d_ws;
  float* xw   = ws;                       // n1*H floats (becomes g1 in place)
  float* agg  = xw + (size_t)n1 * H;      // n2*H
  float* ssum = agg + (size_t)n2 * H;     // n2*H
  float* deg  = ssum + (size_t)n2 * H;    // n2
  float* dinv = deg + n2;                 // n2
  float* cnt  = dinv + n2;                // n2

  // zero agg, ssum, deg, dinv, cnt (contiguous region)
  size_t zn = 2 * (size_t)n2 * H + 3 * (size_t)n2;
  zero_f32<<<1024, 256, 0, stream>>>(agg, zn);

  deg_accum<<<(E1 + 255) / 256, 256, 0, stream>>>(tgt1, e_id1, attr, deg, E1, n2);
  dinv_k<<<(n2 + 255) / 256, 256, 0, stream>>>(deg, dinv, n2);

  xw_gemm<<<n1 / 16, 256, 0, stream>>>(x, gcn_W, xw);

  gcn_scatter<<<(E1 * 32 + 255) / 256, 256, 0, stream>>>(xw, src1, tgt1, e_id1, attr,
                                                         dinv, agg, E1, n2);
  finalize_g1<<<(int)(((size_t)n1 * H + 255) / 256), 256, 0, stream>>>(xw, agg, dinv,
                                                                       gcn_b, n1, n2);
  sage_scatter<<<(E1 * 32 + 255) / 256, 256, 0, stream>>>(xw, src1, tgt1, ssum, cnt,
                                                          E1, n2);
  out_gemm<<<n2 / 16, 256, 0, stream>>>(ssum, cnt, xw, Wl1, bl1, Wr1, (float*)d_out);
  l2norm<<<(n2 * 32 + 255) / 256, 256, 0, stream>>>((float*)d_out, n2);
}